// SelfAttention_63909113364969
// MI455X (gfx1250) — compile-verified
//
#include <hip/hip_runtime.h>
#include <hip/hip_bf16.h>
#include <cmath>

typedef __attribute__((ext_vector_type(2))) float v2f;
typedef __attribute__((ext_vector_type(8))) float v8f;

#define B_ 16
#define S_ 256
#define E_ 32
#define H_ 64

// ---------------------------------------------------------------------------
// Kernel 1: per 16-row tile of flattened (b,s):
//   xn = normalize(x);  a = xn@W1a + b1;  bf = xn@W1b;  v = x@Wv^T + bv
// One wave (32 lanes) per tile, all GEMMs via V_WMMA_F32_16X16X4_F32.
// ---------------------------------------------------------------------------
__global__ __launch_bounds__(32) void proj_kernel(
    const float* __restrict__ x, const float* __restrict__ Wv,
    const float* __restrict__ bv, const float* __restrict__ W1,
    const float* __restrict__ b1,
    float* __restrict__ a_ws, float* __restrict__ bf_ws,
    float* __restrict__ v_ws)
{
  const int tile = blockIdx.x;       // 256 tiles of 16 rows over (b*s)
  const int lane = threadIdx.x;      // 0..31
  const int rc   = lane & 15;        // A-row / B-col index
  const int kh   = lane >> 4;        // K-half selector for 16x16x4 frags
  const int rowbase = tile * 16;

  // Raw-x A-fragments for K=32 in 8 steps of 4; accumulate squared norm.
  v2f xr[8];
  float ss = 0.f;
  const float* xrow = x + (size_t)(rowbase + rc) * E_;
#pragma unroll
  for (int kb = 0; kb < 8; ++kb) {
    const int k0 = kb * 4 + 2 * kh;
    v2f t; t.x = xrow[k0]; t.y = xrow[k0 + 1];
    xr[kb] = t;
    ss += t.x * t.x + t.y * t.y;
  }
  ss += __shfl_xor(ss, 16, 32);                // join the two K-halves of a row
  const float inv = 1.0f / fmaxf(sqrtf(ss), 1e-12f);
  v2f xnr[8];
#pragma unroll
  for (int kb = 0; kb < 8; ++kb) xnr[kb] = xr[kb] * inv;

  // a = xn @ W1[:E] + b1   and   bf = xn @ W1[E:]
#pragma unroll
  for (int half = 0; half < 2; ++half) {
    const int ebase = half * E_;
    float* dst = half ? bf_ws : a_ws;
#pragma unroll
    for (int nt = 0; nt < 4; ++nt) {           // 4 tiles cover H=64
      const int hb = nt * 16;
      v8f acc = {};
#pragma unroll
      for (int kb = 0; kb < 8; ++kb) {
        const int k0 = kb * 4 + 2 * kh;
        v2f bfrag;
        bfrag.x = W1[(size_t)(ebase + k0)     * H_ + hb + rc];
        bfrag.y = W1[(size_t)(ebase + k0 + 1) * H_ + hb + rc];
        acc = __builtin_amdgcn_wmma_f32_16x16x4_f32(
            false, xnr[kb], false, bfrag, (short)0, acc, false, false);
      }
      const float badd = half ? 0.f : b1[hb + rc];
#pragma unroll
      for (int r = 0; r < 8; ++r) {
        const int grow = rowbase + r + 8 * kh;
        dst[(size_t)grow * H_ + hb + rc] = acc[r] + badd;
      }
    }
  }

  // v = x @ Wv^T + bv   (B[k][n] = Wv[n][k])
#pragma unroll
  for (int nt = 0; nt < 2; ++nt) {             // 2 tiles cover E=32
    const int db = nt * 16;
    v8f acc = {};
#pragma unroll
    for (int kb = 0; kb < 8; ++kb) {
      const int k0 = kb * 4 + 2 * kh;
      v2f bfrag;
      bfrag.x = Wv[(size_t)(db + rc) * E_ + k0];
      bfrag.y = Wv[(size_t)(db + rc) * E_ + k0 + 1];
      acc = __builtin_amdgcn_wmma_f32_16x16x4_f32(
          false, xr[kb], false, bfrag, (short)0, acc, false, false);
    }
#pragma unroll
    for (int r = 0; r < 8; ++r) {
      const int grow = rowbase + r + 8 * kh;
      v_ws[(size_t)grow * E_ + db + rc] = acc[r] + bv[db + rc];
    }
  }
}

// ---------------------------------------------------------------------------
// Kernel 2: per (batch, i-tile of 16):
//   scores[i,j] = tanh( w2 . tanh(a_i + bf_j) + b2 );  softmax over j;
//   out tile    = attn(16x256) @ v_b(256x32)  via f32 WMMA (waves 0,1).
// ---------------------------------------------------------------------------
__global__ __launch_bounds__(256) void attn_kernel(
    const float* __restrict__ w2, const float* __restrict__ b2p,
    const float* __restrict__ a_ws, const float* __restrict__ bf_ws,
    const float* __restrict__ v_ws, float* __restrict__ out)
{
  __shared__ float s_a[16 * H_];       // a tile (b1 already folded in)
  __shared__ float s_sc[16 * S_];      // attn tile after softmax
  __shared__ float s_w2[H_];

  const int tid   = threadIdx.x;
  const int b     = blockIdx.x >> 4;
  const int itile = blockIdx.x & 15;
  const int irow0 = b * S_ + itile * 16;

  for (int idx = tid; idx < 16 * H_; idx += 256)
    s_a[idx] = a_ws[(size_t)irow0 * H_ + idx];
  if (tid < H_) s_w2[tid] = w2[tid];
  __syncthreads();

  const float b2 = b2p[0];
  const int i  = tid >> 4;             // 16 rows
  const int jl = tid & 15;             // 16 lanes per row, 16 j's each

  float sc[16];
  float m = -3.0e38f;
  for (int jj = 0; jj < 16; ++jj) {
    const int j = jj * 16 + jl;
    const float* bfr = bf_ws + (size_t)(b * S_ + j) * H_;
    const float* ar  = s_a + i * H_;
    float acc = 0.f;
#pragma unroll 8
    for (int h = 0; h < H_; ++h)
      acc += s_w2[h] * tanhf(ar[h] + bfr[h]);
    const float s = tanhf(acc + b2);
    sc[jj] = s;
    m = fmaxf(m, s);
  }
  // softmax over j: reduce across the 16-lane sub-group owning row i
#pragma unroll
  for (int mask = 1; mask < 16; mask <<= 1)
    m = fmaxf(m, __shfl_xor(m, mask, 16));
  float sum = 0.f;
  for (int jj = 0; jj < 16; ++jj) { sc[jj] = expf(sc[jj] - m); sum += sc[jj]; }
#pragma unroll
  for (int mask = 1; mask < 16; mask <<= 1)
    sum += __shfl_xor(sum, mask, 16);
  const float isum = 1.0f / sum;
  for (int jj = 0; jj < 16; ++jj)
    s_sc[i * S_ + jj * 16 + jl] = sc[jj] * isum;
  __syncthreads();

  // out = attn @ v : waves 0 and 1 each produce one 16x16 d-half, K=256.
  const int wave = tid >> 5;           // wave-uniform branch => EXEC all ones
  if (wave < 2) {
    const int lane = tid & 31;
    const int rc = lane & 15;
    const int kh = lane >> 4;
    const int db = wave * 16;
    const float* vb = v_ws + (size_t)b * S_ * E_;
    v8f acc = {};
    for (int kb = 0; kb < 64; ++kb) {
      const int k0 = kb * 4 + 2 * kh;
      v2f af, bfrag;
      af.x    = s_sc[rc * S_ + k0];
      af.y    = s_sc[rc * S_ + k0 + 1];
      bfrag.x = vb[(size_t)k0       * E_ + db + rc];
      bfrag.y = vb[(size_t)(k0 + 1) * E_ + db + rc];
      acc = __builtin_amdgcn_wmma_f32_16x16x4_f32(
          false, af, false, bfrag, (short)0, acc, false, false);
    }
#pragma unroll
    for (int r = 0; r < 8; ++r) {
      const int grow = irow0 + r + 8 * kh;
      out[(size_t)grow * E_ + db + rc] = acc[r];
    }
  }
}

// ---------------------------------------------------------------------------
extern "C" void kernel_launch(void* const* d_in, const int* in_sizes, int n_in,
                              void* d_out, int out_size, void* d_ws, size_t ws_size,
                              hipStream_t stream) {
  const float* x  = (const float*)d_in[0];
  const float* Wv = (const float*)d_in[1];
  const float* bv = (const float*)d_in[2];
  const float* W1 = (const float*)d_in[3];
  const float* b1 = (const float*)d_in[4];
  const float* w2 = (const float*)d_in[5];
  const float* b2 = (const float*)d_in[6];
  float* outp = (float*)d_out;

  float* ws    = (float*)d_ws;
  float* a_ws  = ws;                                   // B*S*H
  float* bf_ws = a_ws + (size_t)B_ * S_ * H_;          // B*S*H
  float* v_ws  = bf_ws + (size_t)B_ * S_ * H_;         // B*S*E

  proj_kernel<<<dim3(B_ * S_ / 16), dim3(32), 0, stream>>>(
      x, Wv, bv, W1, b1, a_ws, bf_ws, v_ws);
  attn_kernel<<<dim3(B_ * (S_ / 16)), dim3(256), 0, stream>>>(
      w2, b2, a_ws, bf_ws, v_ws, outp);
}